// GraphVAEEncoder_56934086476461
// MI455X (gfx1250) — compile-verified
//
#include <hip/hip_runtime.h>

typedef __attribute__((ext_vector_type(2))) float v2f;
typedef __attribute__((ext_vector_type(8))) float v8f;

#define N_NODES_C   100000
#define N_EDGES_C   3200000
#define N_GRAPHS_C  64
#define IN_DIM_C    128
#define HIDDEN_C    64
#define LATENT_C    16
#define BPAIR_STRIDE 160   // floats per k-pair row in LDS (padded: lanes16-31 -> banks 32-63)

#ifndef __has_builtin
#define __has_builtin(x) 0
#endif

#if __has_builtin(__builtin_amdgcn_global_load_async_to_lds_b128)
#define HAVE_ASYNC_LDS 1
#else
#define HAVE_ASYNC_LDS 0
#endif

#if __has_builtin(__builtin_amdgcn_s_wait_asynccnt)
#define WAIT_ASYNC() __builtin_amdgcn_s_wait_asynccnt(0)
#else
#define WAIT_ASYNC() asm volatile("s_wait_asynccnt 0x0" ::: "memory")
#endif

// 16-byte int vector, matching the builtin's expected pointee type
typedef int v4i_t __attribute__((vector_size(4 * sizeof(int))));
typedef __attribute__((address_space(1))) v4i_t* gv4p_t;   // global (AS1)
typedef __attribute__((address_space(3))) v4i_t* lv4p_t;   // LDS (AS3)

// ---------------------------------------------------------------------------
// zero fill
// ---------------------------------------------------------------------------
__global__ void zero_kernel(float* __restrict__ p, long long n) {
    long long i = (long long)blockIdx.x * blockDim.x + threadIdx.x;
    if (i < n) p[i] = 0.0f;
}

// ---------------------------------------------------------------------------
// degree accumulation: deg_out[src[e]] += 1, deg_in[dst[e]] += 1
// ---------------------------------------------------------------------------
__global__ void degree_kernel(const int* __restrict__ src, const int* __restrict__ dst,
                              float* __restrict__ degO, float* __restrict__ degI, int nE) {
    int e = blockIdx.x * blockDim.x + threadIdx.x;
    if (e < nE) {
        atomicAdd(&degO[src[e]], 1.0f);
        atomicAdd(&degI[dst[e]], 1.0f);
    }
}

// deg -> clip(deg,1)^-0.5 in place
__global__ void rsqrt_clip_kernel(float* __restrict__ p, int n) {
    int i = blockIdx.x * blockDim.x + threadIdx.x;
    if (i < n) {
        float d = p[i];
        d = d < 1.0f ? 1.0f : d;
        p[i] = 1.0f / sqrtf(d);
    }
}

// ---------------------------------------------------------------------------
// out[row, 0:64] = (X[row, 0:K] @ W[K,64]) * scale[row]
// One wave per 16-row tile; wave computes all four 16-col tiles of the
// 64-wide output via V_WMMA_F32_16X16X4_F32 accumulation over K.
//
// W staging: async-copy (GLOBAL_LOAD_ASYNC_TO_LDS_B128, ASYNCcnt) into a raw
// LDS buffer, then repacked into a k-pair-interleaved layout so each WMMA B
// operand is a single contiguous ds_load_b64 with no cross-register shuffles
// and no bank conflicts.
//
// VGPR layouts per CDNA5 ISA 7.12.2:
//   A (16x4 f32, 2 VGPRs): lanes 0-15 hold M=lane, v0=K0,v1=K1;
//                          lanes 16-31 hold M=lane-16, v0=K2,v1=K3.
//   B (4x16 f32, 2 VGPRs): lanes 0-15 hold N=lane, v0=K0,v1=K1;
//                          lanes 16-31 hold N=lane-16, v0=K2,v1=K3.
//   C/D (16x16 f32, 8 VGPRs): vgpr r: lanes 0-15 M=r, lanes 16-31 M=r+8.
// ---------------------------------------------------------------------------
__global__ void __launch_bounds__(256)
gemm_scale_wmma(const float* __restrict__ X, const float* __restrict__ Wg,
                const float* __restrict__ scale, float* __restrict__ out,
                int N, int K) {
    __shared__ __align__(16) float ldsRaw[IN_DIM_C * HIDDEN_C];          // <= 32 KB
    __shared__ __align__(16) float ldsB[(IN_DIM_C / 2) * BPAIR_STRIDE];  // 40 KB

    // Stage raw W into LDS
#if HAVE_ASYNC_LDS
    {
        gv4p_t gsrc = (gv4p_t)(unsigned long long)(const void*)Wg;
        lv4p_t ldst = (lv4p_t)(unsigned long long)(void*)ldsRaw;
        const int nvec = (K * HIDDEN_C) / 4;   // 16-byte chunks
        for (int v = threadIdx.x; v < nvec; v += blockDim.x)
            __builtin_amdgcn_global_load_async_to_lds_b128(gsrc + v, ldst + v, 0, 0);
        WAIT_ASYNC();
    }
#else
    for (int i = threadIdx.x; i < K * HIDDEN_C; i += blockDim.x)
        ldsRaw[i] = Wg[i];
#endif
    __syncthreads();

    // Repack: ldsB[(kk/2)*STRIDE + col*2 + (kk&1)] = W[kk][col]
    for (int i = threadIdx.x; i < K * HIDDEN_C; i += blockDim.x) {
        int kk  = i >> 6;
        int col = i & 63;
        ldsB[(kk >> 1) * BPAIR_STRIDE + (col << 1) + (kk & 1)] = ldsRaw[i];
    }
    __syncthreads();

    const int lane  = threadIdx.x & 31;
    const int wave  = threadIdx.x >> 5;
    const int tile  = blockIdx.x * 8 + wave;      // 16-row tile index
    const int row0  = tile * 16;
    if (row0 >= N) return;                        // wave-uniform exit

    const int mrow  = lane & 15;
    const int khalf = lane >> 4;                  // 0 or 1

    v8f acc0 = {0.f,0.f,0.f,0.f,0.f,0.f,0.f,0.f};
    v8f acc1 = {0.f,0.f,0.f,0.f,0.f,0.f,0.f,0.f};
    v8f acc2 = {0.f,0.f,0.f,0.f,0.f,0.f,0.f,0.f};
    v8f acc3 = {0.f,0.f,0.f,0.f,0.f,0.f,0.f,0.f};

    const float* xrow = X + (size_t)(row0 + mrow) * K + 2 * khalf;

    for (int k = 0; k < K; k += 4) {
        v2f a;
        a.x = xrow[k];
        a.y = xrow[k + 1];

        // B operands: one contiguous 8-byte LDS load each
        const int bbase = ((k >> 1) + khalf) * BPAIR_STRIDE + (mrow << 1);
        const v2f b0 = *(const v2f*)&ldsB[bbase];
        const v2f b1 = *(const v2f*)&ldsB[bbase + 32];
        const v2f b2 = *(const v2f*)&ldsB[bbase + 64];
        const v2f b3 = *(const v2f*)&ldsB[bbase + 96];

        acc0 = __builtin_amdgcn_wmma_f32_16x16x4_f32(false, a, false, b0, (short)0, acc0, false, false);
        acc1 = __builtin_amdgcn_wmma_f32_16x16x4_f32(false, a, false, b1, (short)0, acc1, false, false);
        acc2 = __builtin_amdgcn_wmma_f32_16x16x4_f32(false, a, false, b2, (short)0, acc2, false, false);
        acc3 = __builtin_amdgcn_wmma_f32_16x16x4_f32(false, a, false, b3, (short)0, acc3, false, false);
    }

    // D layout: element r in this lane is row (row0 + r + 8*khalf), col (ctile*16 + mrow)
#pragma unroll
    for (int r = 0; r < 8; ++r) {
        int row = row0 + r + 8 * khalf;
        if (row < N) {
            float sc = scale[row];
            float* orow = out + (size_t)row * HIDDEN_C + mrow;
            orow[0]  = acc0[r] * sc;
            orow[16] = acc1[r] * sc;
            orow[32] = acc2[r] * sc;
            orow[48] = acc3[r] * sc;
        }
    }
}

// ---------------------------------------------------------------------------
// agg[dst[e], c] += hw[src[e], c]   (one thread per edge-channel; L2-resident)
// ---------------------------------------------------------------------------
__global__ void scatter_add_kernel(const float* __restrict__ hw, const int* __restrict__ src,
                                   const int* __restrict__ dst, float* __restrict__ agg,
                                   int nE) {
    long long idx = (long long)blockIdx.x * blockDim.x + threadIdx.x;
    int e = (int)(idx >> 6);
    int c = (int)(idx & 63);
    if (e < nE) {
        int s = src[e];
        int d = dst[e];
        float v = hw[(size_t)s * HIDDEN_C + c];
        atomicAdd(&agg[(size_t)d * HIDDEN_C + c], v);
    }
}

// h[i,c] = relu(agg[i,c] * deg_in_inv[i] + bias[c])
__global__ void finalize_relu_kernel(const float* __restrict__ agg, const float* __restrict__ degI,
                                     const float* __restrict__ bias, float* __restrict__ h, int nN) {
    long long idx = (long long)blockIdx.x * blockDim.x + threadIdx.x;
    int i = (int)(idx >> 6);
    int c = (int)(idx & 63);
    if (i < nN) {
        float v = agg[(size_t)i * HIDDEN_C + c] * degI[i] + bias[c];
        h[(size_t)i * HIDDEN_C + c] = v > 0.0f ? v : 0.0f;
    }
}

// final layer: relu + accumulate into per-graph sums
__global__ void finalize_graphsum_kernel(const float* __restrict__ agg, const float* __restrict__ degI,
                                         const float* __restrict__ bias, const int* __restrict__ node_graph,
                                         float* __restrict__ gsum, int nN) {
    long long idx = (long long)blockIdx.x * blockDim.x + threadIdx.x;
    int i = (int)(idx >> 6);
    int c = (int)(idx & 63);
    if (i < nN) {
        float v = agg[(size_t)i * HIDDEN_C + c] * degI[i] + bias[c];
        v = v > 0.0f ? v : 0.0f;
        atomicAdd(&gsum[(size_t)node_graph[i] * HIDDEN_C + c], v);
    }
}

__global__ void count_kernel(const int* __restrict__ node_graph, float* __restrict__ gcount, int nN) {
    int i = blockIdx.x * blockDim.x + threadIdx.x;
    if (i < nN) atomicAdd(&gcount[node_graph[i]], 1.0f);
}

// mu/logvar heads: out[half*1024 + g*16 + lat]
__global__ void head_kernel(const float* __restrict__ gsum, const float* __restrict__ gcount,
                            const float* __restrict__ Wmu, const float* __restrict__ bmu,
                            const float* __restrict__ Wlv, const float* __restrict__ blv,
                            float* __restrict__ out) {
    int t = blockIdx.x * blockDim.x + threadIdx.x;
    if (t >= N_GRAPHS_C * LATENT_C * 2) return;
    int half = t >= N_GRAPHS_C * LATENT_C;
    int r = t - half * N_GRAPHS_C * LATENT_C;
    int g   = r / LATENT_C;
    int lat = r % LATENT_C;
    float cnt = gcount[g];
    cnt = cnt < 1.0f ? 1.0f : cnt;
    float inv = 1.0f / cnt;
    const float* Wsel = half ? Wlv : Wmu;
    float bsel = half ? blv[lat] : bmu[lat];
    float s = 0.0f;
    for (int k = 0; k < HIDDEN_C; ++k)
        s += gsum[g * HIDDEN_C + k] * inv * Wsel[k * LATENT_C + lat];
    out[t] = s + bsel;
}

// ---------------------------------------------------------------------------
extern "C" void kernel_launch(void* const* d_in, const int* in_sizes, int n_in,
                              void* d_out, int out_size, void* d_ws, size_t ws_size,
                              hipStream_t stream) {
    (void)in_sizes; (void)n_in; (void)out_size; (void)ws_size;

    const float* x   = (const float*)d_in[0];
    // d_in[1] edge_feat: does not affect (mu, logvar)
    const int*   src = (const int*)d_in[2];
    const int*   dst = (const int*)d_in[3];
    const int*   ng  = (const int*)d_in[4];
    const float* W1  = (const float*)d_in[5];
    const float* b1  = (const float*)d_in[6];
    const float* W2  = (const float*)d_in[7];
    const float* b2  = (const float*)d_in[8];
    const float* Wmu = (const float*)d_in[11];
    const float* bmu = (const float*)d_in[12];
    const float* Wlv = (const float*)d_in[13];
    const float* blv = (const float*)d_in[14];
    float* out = (float*)d_out;

    float* ws   = (float*)d_ws;
    float* degO = ws;                                   // N
    float* degI = degO + N_NODES_C;                     // N
    float* B0   = degI + N_NODES_C;                     // N*64
    float* B1   = B0 + (size_t)N_NODES_C * HIDDEN_C;    // N*64
    float* gsum = B1 + (size_t)N_NODES_C * HIDDEN_C;    // 64*64
    float* gcnt = gsum + N_GRAPHS_C * HIDDEN_C;         // 64

    const long long nodeElems = (long long)N_NODES_C * HIDDEN_C;
    const long long edgeElems = (long long)N_EDGES_C * HIDDEN_C;

    auto zero = [&](float* p, long long n) {
        int blocks = (int)((n + 255) / 256);
        zero_kernel<<<blocks, 256, 0, stream>>>(p, n);
    };

    zero(degO, N_NODES_C);
    zero(degI, N_NODES_C);
    zero(B1, nodeElems);
    zero(gsum, N_GRAPHS_C * HIDDEN_C);
    zero(gcnt, N_GRAPHS_C);

    degree_kernel<<<(N_EDGES_C + 255) / 256, 256, 0, stream>>>(src, dst, degO, degI, N_EDGES_C);
    rsqrt_clip_kernel<<<(N_NODES_C + 255) / 256, 256, 0, stream>>>(degO, N_NODES_C);
    rsqrt_clip_kernel<<<(N_NODES_C + 255) / 256, 256, 0, stream>>>(degI, N_NODES_C);

    const int tiles   = (N_NODES_C + 15) / 16;   // 6250
    const int gblocks = (tiles + 7) / 8;         // 8 waves/block, 1 tile/wave

    // Layer 1: hw = (x @ W1) * deg_out^-0.5  -> B0
    gemm_scale_wmma<<<gblocks, 256, 0, stream>>>(x, W1, degO, B0, N_NODES_C, IN_DIM_C);
    // agg = scatter_add(hw[src] -> dst)      -> B1
    scatter_add_kernel<<<(int)((edgeElems + 255) / 256), 256, 0, stream>>>(B0, src, dst, B1, N_EDGES_C);
    // h = relu(agg * deg_in^-0.5 + b1)       -> B0
    finalize_relu_kernel<<<(int)((nodeElems + 255) / 256), 256, 0, stream>>>(B1, degI, b1, B0, N_NODES_C);

    // Layer 2: hw2 = (h @ W2) * deg_out^-0.5 -> B1
    gemm_scale_wmma<<<gblocks, 256, 0, stream>>>(B0, W2, degO, B1, N_NODES_C, HIDDEN_C);
    zero(B0, nodeElems);
    scatter_add_kernel<<<(int)((edgeElems + 255) / 256), 256, 0, stream>>>(B1, src, dst, B0, N_EDGES_C);

    // Graph mean + heads
    count_kernel<<<(N_NODES_C + 255) / 256, 256, 0, stream>>>(ng, gcnt, N_NODES_C);
    finalize_graphsum_kernel<<<(int)((nodeElems + 255) / 256), 256, 0, stream>>>(B0, degI, b2, ng, gsum, N_NODES_C);
    head_kernel<<<8, 256, 0, stream>>>(gsum, gcnt, Wmu, bmu, Wlv, blv, out);
}